// OptNetLayer_69561290326601
// MI455X (gfx1250) — compile-verified
//
#include <hip/hip_runtime.h>

// CDNA5 (gfx1250) wave32 WMMA FP32 path: V_WMMA_F32_16X16X4_F32
typedef float v2f __attribute__((ext_vector_type(2)));
typedef float v8f __attribute__((ext_vector_type(8)));

#define NN      64          // n_z == n_ineq == 64
#define LDH     65          // LDS row stride (stride-65 -> conflict-free columns)
#define NITER   20
#define SIGMA_C 0.1f
#define BIGV    1e8f

__launch_bounds__(128)
__global__ void optnet_ipm_wmma(const float* __restrict__ P,
                                const float* __restrict__ Q,
                                const float* __restrict__ G,
                                const float* __restrict__ H,
                                float* __restrict__ OUT)
{
    __shared__ float Qs[NN * LDH];
    __shared__ float Gs[NN * LDH];
    __shared__ float Hs[NN * LDH];           // KKT matrix -> in-place augmented LU
    __shared__ float vz[NN], vs[NN], vl[NN]; // z, s, lambda
    __shared__ float dd[NN];                 // lambda/s
    __shared__ float tt[NN];                 // (lam*r_prim - r_comp)/s
    __shared__ float bb[NN];                 // rhs (augmented column of LU)
    __shared__ float dzv[NN];                // dz solution
    __shared__ float red[NN];                // reduction scratch
    __shared__ float part[3][128];           // split-matvec partial sums
    __shared__ float mu_s, alpha_s;

    const int tid  = threadIdx.x;
    const int lane = tid & 31;
    const int wave = tid >> 5;               // 4 waves: wave == WMMA tile-row ti
    const int b    = blockIdx.x;

    // ---- stage shared matrices into LDS ----
    for (int idx = tid; idx < NN * NN; idx += 128) {
        const int r = idx >> 6, c = idx & 63;
        Qs[r * LDH + c] = Q[idx];
        Gs[r * LDH + c] = G[idx];
    }
    float p_i = 0.f, h_i = 0.f;
    if (tid < NN) {
        p_i = P[b * NN + tid];
        h_i = H[tid];
        vz[tid] = 0.f; vs[tid] = 1.f; vl[tid] = 1.f;
    }
    __syncthreads();

    // WMMA fragment index helpers (ISA 16x16x4 f32 layouts, wave32)
    const int fm  = lane & 15;               // M (A) / N (B,C) index
    const int kb  = (lane >> 4) * 2;         // K base: lanes 0-15 -> k{0,1}, 16-31 -> k{2,3}
    const int hi8 = (lane >> 4) * 8;         // C/D: lanes 16-31 hold rows M+8

    // ---- iteration-invariant register tiles ----
    // Q tiles (C seed) for this wave's tile-row
    v8f qacc0, qacc1, qacc2, qacc3;
    #pragma unroll
    for (int v = 0; v < 8; ++v) {
        const int row = (wave * 16 + hi8 + v) * LDH;
        qacc0[v] = Qs[row +  0 + fm];
        qacc1[v] = Qs[row + 16 + fm];
        qacc2[v] = Qs[row + 32 + fm];
        qacc3[v] = Qs[row + 48 + fm];
    }
    // A-base fragments: G' tile for this wave's tile-row, all 16 K-slabs
    v2f abase[16];
    #pragma unroll
    for (int kk = 0; kk < 16; ++kk) {
        const int k0 = kk * 4 + kb;
        abase[kk].x = Gs[k0 * LDH + wave * 16 + fm];
        abase[kk].y = Gs[(k0 + 1) * LDH + wave * 16 + fm];
    }

    // LU 2-D tiling coordinates (8 row-groups x 16 col-groups)
    const int ri = tid >> 4;                 // 0..7
    const int cj = tid & 15;                 // 0..15

    float rd_i = 0.f, rp_i = 0.f, rc_i = 0.f;

    for (int it = 0; it < NITER; ++it) {
        // ---- residual matvecs, split over j-halves across all 128 threads ----
        {
            const int ia = tid & 63;
            const int j0 = (tid >> 6) * 32;
            float qz = 0.f, gz = 0.f, gtl = 0.f;
            for (int j = j0; j < j0 + 32; ++j) {
                const float zj = vz[j];
                qz  += Qs[ia * LDH + j] * zj;
                gz  += Gs[ia * LDH + j] * zj;
                gtl += Gs[j * LDH + ia] * vl[j];
            }
            part[0][tid] = qz; part[1][tid] = gz; part[2][tid] = gtl;
        }
        __syncthreads();
        if (tid < NN) {
            const float qz  = part[0][tid] + part[0][tid + 64];
            const float gz  = part[1][tid] + part[1][tid + 64];
            const float gtl = part[2][tid] + part[2][tid + 64];
            const float si = vs[tid], li = vl[tid];
            rd_i = qz + p_i + gtl;           // Qz + p + G'lam
            rp_i = gz + si - h_i;            // Gz + s - h
            red[tid] = si * li;
        }
        __syncthreads();
        // ---- mu = dot(s,lam)/m ----
        if (wave == 0) {
            float v = red[lane] + red[lane + 32];
            for (int off = 16; off > 0; off >>= 1) v += __shfl_xor(v, off, 32);
            if (lane == 0) mu_s = v * (1.0f / NN);
        }
        __syncthreads();
        // ---- r_comp, t, d ----
        if (tid < NN) {
            const float si = vs[tid], li = vl[tid];
            rc_i = si * li - SIGMA_C * mu_s;
            tt[tid] = (li * rp_i - rc_i) / si;
            dd[tid] = li / si;
        }
        __syncthreads();
        // ---- rhs = -(r_dual + G' t), split matvec ----
        {
            const int ia = tid & 63;
            const int j0 = (tid >> 6) * 32;
            float gt = 0.f;
            for (int j = j0; j < j0 + 32; ++j) gt += Gs[j * LDH + ia] * tt[j];
            part[0][tid] = gt;
        }
        __syncthreads();
        if (tid < NN) bb[tid] = -(rd_i + part[0][tid] + part[0][tid + 64]);
        __syncthreads();

        // ---- H = Q + (G'D) G via FP32 WMMA; scale folded into A ----
        {
            v8f acc0 = qacc0, acc1 = qacc1, acc2 = qacc2, acc3 = qacc3;
            #pragma unroll
            for (int kk = 0; kk < 16; ++kk) {
                const int k0 = kk * 4 + kb, k1 = k0 + 1;
                const float d0 = dd[k0], d1 = dd[k1];
                v2f a;                                   // A = (G'D) tile
                a.x = d0 * abase[kk].x;
                a.y = d1 * abase[kk].y;
                v2f b0, b1, b2, b3;                      // B = plain G tiles
                b0.x = Gs[k0 * LDH +  0 + fm];  b0.y = Gs[k1 * LDH +  0 + fm];
                b1.x = Gs[k0 * LDH + 16 + fm];  b1.y = Gs[k1 * LDH + 16 + fm];
                b2.x = Gs[k0 * LDH + 32 + fm];  b2.y = Gs[k1 * LDH + 32 + fm];
                b3.x = Gs[k0 * LDH + 48 + fm];  b3.y = Gs[k1 * LDH + 48 + fm];
                acc0 = __builtin_amdgcn_wmma_f32_16x16x4_f32(false, a, false, b0, (short)0, acc0, false, false);
                acc1 = __builtin_amdgcn_wmma_f32_16x16x4_f32(false, a, false, b1, (short)0, acc1, false, false);
                acc2 = __builtin_amdgcn_wmma_f32_16x16x4_f32(false, a, false, b2, (short)0, acc2, false, false);
                acc3 = __builtin_amdgcn_wmma_f32_16x16x4_f32(false, a, false, b3, (short)0, acc3, false, false);
            }
            #pragma unroll
            for (int v = 0; v < 8; ++v) {
                const int row = (wave * 16 + hi8 + v) * LDH;
                Hs[row +  0 + fm] = acc0[v];
                Hs[row + 16 + fm] = acc1[v];
                Hs[row + 32 + fm] = acc2[v];
                Hs[row + 48 + fm] = acc3[v];
            }
        }
        __syncthreads();

        // ---- augmented Gaussian elimination on [H | b], 2-D tiled ----
        // L is never stored (not needed); column k never written -> race-free,
        // one barrier per step. b becomes the forward-solved vector y.
        for (int k = 0; k < NN - 1; ++k) {
            const float invpk = 1.0f / Hs[k * LDH + k];
            const float bk = bb[k];
            const int jb = k + 1 + cj;
            float hk0 = 0.f, hk1 = 0.f, hk2 = 0.f, hk3 = 0.f;
            if (jb      < NN) hk0 = Hs[k * LDH + jb];
            if (jb + 16 < NN) hk1 = Hs[k * LDH + jb + 16];
            if (jb + 32 < NN) hk2 = Hs[k * LDH + jb + 32];
            if (jb + 48 < NN) hk3 = Hs[k * LDH + jb + 48];
            for (int i = k + 1 + ri; i < NN; i += 8) {
                const float f = Hs[i * LDH + k] * invpk;
                if (jb      < NN) Hs[i * LDH + jb]      -= f * hk0;
                if (jb + 16 < NN) Hs[i * LDH + jb + 16] -= f * hk1;
                if (jb + 32 < NN) Hs[i * LDH + jb + 32] -= f * hk2;
                if (jb + 48 < NN) Hs[i * LDH + jb + 48] -= f * hk3;
                if (cj == 0) bb[i] -= f * bk;
            }
            __syncthreads();
        }
        // ---- backward substitution (U x = y), solution -> dzv ----
        for (int k = NN - 1; k >= 0; --k) {
            const float xk = bb[k] / Hs[k * LDH + k];
            if (tid == 0) dzv[k] = xk;
            if (tid < k)  bb[tid] -= Hs[tid * LDH + k] * xk;
            __syncthreads();
        }

        // ---- G*dz (split matvec), then ds, dlam, step length ----
        {
            const int ia = tid & 63;
            const int j0 = (tid >> 6) * 32;
            float gdz = 0.f;
            for (int j = j0; j < j0 + 32; ++j) gdz += Gs[ia * LDH + j] * dzv[j];
            part[0][tid] = gdz;
        }
        __syncthreads();
        float ds_i = 0.f, dl_i = 0.f;
        if (tid < NN) {
            const float gdz = part[0][tid] + part[0][tid + 64];
            ds_i = -rp_i - gdz;
            const float si = vs[tid], li = vl[tid];
            dl_i = (-rc_i - li * ds_i) / si;
            const float c1 = (ds_i < 0.f) ? (-si / ds_i) : BIGV;
            const float c2 = (dl_i < 0.f) ? (-li / dl_i) : BIGV;
            red[tid] = fminf(c1, c2);
        }
        __syncthreads();
        if (wave == 0) {
            float v = fminf(red[lane], red[lane + 32]);
            for (int off = 16; off > 0; off >>= 1) v = fminf(v, __shfl_xor(v, off, 32));
            if (lane == 0) alpha_s = fminf(1.0f, 0.99f * v);
        }
        __syncthreads();
        if (tid < NN) {
            const float al = alpha_s;
            vz[tid] += al * dzv[tid];
            vs[tid] += al * ds_i;
            vl[tid] += al * dl_i;
        }
        __syncthreads();
    }

    if (tid < NN) OUT[b * NN + tid] = vz[tid];
}

extern "C" void kernel_launch(void* const* d_in, const int* in_sizes, int n_in,
                              void* d_out, int out_size, void* d_ws, size_t ws_size,
                              hipStream_t stream) {
    const float* p = (const float*)d_in[0];   // (B, 64)
    const float* Q = (const float*)d_in[1];   // (64, 64)
    const float* G = (const float*)d_in[2];   // (64, 64)
    const float* h = (const float*)d_in[3];   // (64,)
    float* out = (float*)d_out;               // (B, 64)
    const int B = in_sizes[0] / NN;
    optnet_ipm_wmma<<<dim3(B), dim3(128), 0, stream>>>(p, Q, G, h, out);
}